// TransformerConvNet_4604204941624
// MI455X (gfx1250) — compile-verified
//
#include <hip/hip_runtime.h>
#include <math.h>

// ---------------- problem constants ----------------
#define NB       8
#define NPG      8192
#define NN       (NB * NPG)      // 65536 nodes
#define NE       262144          // edges
#define ND       36              // node dim (35 feats + entity)
#define L1IN     52              // 35 + 16 + 1
#define EH       128             // embed hidden
#define HH       4               // heads
#define CC       64              // per-head channels
#define HCC      256             // HH*CC

typedef float v2f __attribute__((ext_vector_type(2)));
typedef float v8f __attribute__((ext_vector_type(8)));

// fp32 WMMA: D(16x16) = A(16x4) * B(4x16) + C   -- exact fp32 matrix path on CDNA5
__device__ __forceinline__ v8f wmma4(v2f a, v2f b, v8f c) {
  return __builtin_amdgcn_wmma_f32_16x16x4_f32(false, a, false, b, (short)0, c, false, false);
}

__device__ __forceinline__ void atomicMaxF32(float* addr, float v) {
  // sign-aware int-punning trick (works for all sign mixes, init = -inf)
  if (v >= 0.0f) atomicMax((int*)addr, __float_as_int(v));
  else           atomicMin((unsigned int*)addr, __float_as_uint(v));
}

// ---------------- utility kernels ----------------
__global__ void fill_kernel(float* __restrict__ p, float val, int n) {
  int i = blockIdx.x * 256 + threadIdx.x;
  if (i < n) p[i] = val;
}

__global__ void relu_inplace(float* __restrict__ p, int n) {
  int i = blockIdx.x * 256 + threadIdx.x;
  if (i < n) p[i] = fmaxf(p[i], 0.0f);
}

__global__ void gather_out(const float* __restrict__ h, const int* __restrict__ agent,
                           float* __restrict__ out) {
  int b = blockIdx.x, c = threadIdx.x;           // 8 x 256
  int node = b * NPG + agent[b];
  out[b * HCC + c] = fmaxf(h[(size_t)node * HCC + c], 0.0f);
}

// ---------------- edge MLP: 64 edges/block, 4 waves, WMMA f32 ----------------
#define EPB 64
__global__ __launch_bounds__(128) void edge_mlp(
    const float* __restrict__ x, const int* __restrict__ esrc, const int* __restrict__ edst,
    const float* __restrict__ attr, const float* __restrict__ emb,
    const float* __restrict__ w1, const float* __restrict__ b1,
    const float* __restrict__ wh, const float* __restrict__ bh,
    float* __restrict__ h0)
{
  __shared__ int   ssrc[EPB], sdst[EPB], sent[EPB];
  __shared__ float sattr[EPB];
  __shared__ __align__(16) float A1[EPB * 52];    // 16 edges x 52 per wave tile
  __shared__ __align__(16) float A2[EPB * 132];   // hidden, padded stride vs 64 banks

  const int e0 = blockIdx.x * EPB;
  for (int i = threadIdx.x; i < EPB; i += 128) {
    int s = esrc[e0 + i];
    ssrc[i] = s;
    sdst[i] = edst[e0 + i];
    sent[i] = (int)x[s * ND + (ND - 1)];
    sattr[i] = attr[e0 + i];
  }
  __syncthreads();
  for (int i = threadIdx.x; i < EPB * 52; i += 128) {
    int e = i / 52, c = i - e * 52;
    float v;
    if (c < 35)      v = x[ssrc[e] * ND + c];
    else if (c < 51) v = emb[sent[e] * 16 + (c - 35)];
    else             v = sattr[e];
    A1[e * 52 + c] = v;
  }
  __syncthreads();

  const int lane = threadIdx.x & 31;
  const int wave = threadIdx.x >> 5;
  const int hl   = lane >> 4;       // half-wave select
  const int l15  = lane & 15;
  const int koff = hl * 2;
  const int m0   = wave * 16;       // this wave's 16 edges

  // GEMM1: 16x52 @ 52x128 + b1, relu -> A2
  {
    v8f acc[8];
#pragma unroll
    for (int t = 0; t < 8; ++t) {
      float bv = b1[t * 16 + l15];
#pragma unroll
      for (int r = 0; r < 8; ++r) acc[t][r] = bv;
    }
    for (int k0 = 0; k0 < 52; k0 += 4) {
      v2f a = *(const v2f*)&A1[(m0 + l15) * 52 + k0 + koff];
#pragma unroll
      for (int t = 0; t < 8; ++t) {
        int n = t * 16 + l15;
        v2f b;
        b.x = w1[(k0 + koff) * EH + n];
        b.y = w1[(k0 + koff + 1) * EH + n];
        acc[t] = wmma4(a, b, acc[t]);
      }
    }
#pragma unroll
    for (int t = 0; t < 8; ++t) {
      int n = t * 16 + l15;
#pragma unroll
      for (int r = 0; r < 8; ++r)
        A2[(m0 + r + hl * 8) * 132 + n] = fmaxf(acc[t][r], 0.0f);
    }
  }
  __syncthreads();

  // GEMM2: 16x128 @ 128x128 + bh, relu, scatter-add to h0[dst]
  {
    v8f acc[8];
#pragma unroll
    for (int t = 0; t < 8; ++t) {
      float bv = bh[t * 16 + l15];
#pragma unroll
      for (int r = 0; r < 8; ++r) acc[t][r] = bv;
    }
    for (int k0 = 0; k0 < EH; k0 += 4) {
      v2f a = *(const v2f*)&A2[(m0 + l15) * 132 + k0 + koff];
#pragma unroll
      for (int t = 0; t < 8; ++t) {
        int n = t * 16 + l15;
        v2f b;
        b.x = wh[(k0 + koff) * EH + n];
        b.y = wh[(k0 + koff + 1) * EH + n];
        acc[t] = wmma4(a, b, acc[t]);
      }
    }
#pragma unroll
    for (int t = 0; t < 8; ++t) {
      int n = t * 16 + l15;
#pragma unroll
      for (int r = 0; r < 8; ++r) {
        int row = m0 + r + hl * 8;
        atomicAdd(&h0[(size_t)sdst[row] * EH + n], fmaxf(acc[t][r], 0.0f));
      }
    }
  }
}

// ---------------- fused q/k/v/skip node GEMMs, WMMA f32 ----------------
__global__ __launch_bounds__(128) void node_qkvs(
    const float* __restrict__ hin, int K,
    const float* __restrict__ qw, const float* __restrict__ qb,
    const float* __restrict__ kw, const float* __restrict__ kb,
    const float* __restrict__ vw, const float* __restrict__ vb,
    const float* __restrict__ sw, const float* __restrict__ sb,
    float* __restrict__ qo, float* __restrict__ ko,
    float* __restrict__ vo, float* __restrict__ so)
{
  __shared__ __align__(16) float As[16 * 260];   // up to K=256, stride K+4
  const int SA = K + 4;
  const int row0 = blockIdx.x * 16;

  for (int i = threadIdx.x; i < 16 * K; i += 128) {
    int r = i / K, c = i - r * K;
    As[r * SA + c] = hin[(size_t)(row0 + r) * K + c];
  }
  __syncthreads();

  const int lane = threadIdx.x & 31;
  const int wave = threadIdx.x >> 5;
  const int hl   = lane >> 4;
  const int l15  = lane & 15;
  const int koff = hl * 2;
  const int n0   = wave * 64;       // this wave's 64-column slice

  const float* Wm[4] = {qw, kw, vw, sw};
  const float* Bm[4] = {qb, kb, vb, sb};
  float*       Om[4] = {qo, ko, vo, so};

#pragma unroll
  for (int m = 0; m < 4; ++m) {
    const float* W = Wm[m];
    const float* bias = Bm[m];
    float* out = Om[m];
    v8f acc[4];
#pragma unroll
    for (int t = 0; t < 4; ++t) {
      float bv = bias[n0 + t * 16 + l15];
#pragma unroll
      for (int r = 0; r < 8; ++r) acc[t][r] = bv;
    }
    for (int k0 = 0; k0 < K; k0 += 4) {
      v2f a = *(const v2f*)&As[l15 * SA + k0 + koff];
#pragma unroll
      for (int t = 0; t < 4; ++t) {
        int n = n0 + t * 16 + l15;
        v2f b;
        b.x = W[(k0 + koff) * HCC + n];
        b.y = W[(k0 + koff + 1) * HCC + n];
        acc[t] = wmma4(a, b, acc[t]);
      }
    }
#pragma unroll
    for (int t = 0; t < 4; ++t) {
      int n = n0 + t * 16 + l15;
#pragma unroll
      for (int r = 0; r < 8; ++r)
        out[(size_t)(row0 + r + hl * 8) * HCC + n] = acc[t][r];
    }
  }
}

// ---------------- attention: alpha + running max (wave per edge) ----------------
__global__ __launch_bounds__(128) void attn_alpha(
    const float* __restrict__ q, const float* __restrict__ k,
    const int* __restrict__ esrc, const int* __restrict__ edst,
    const float* __restrict__ attr, const float* __restrict__ ew,
    float* __restrict__ ealpha, float* __restrict__ amax)
{
  const int e = blockIdx.x * 4 + (threadIdx.x >> 5);
  const int lane = threadIdx.x & 31;
  const int s = esrc[e], d = edst[e];
  const float a = attr[e];

  const float4* qp = (const float4*)(q + (size_t)d * HCC + lane * 8);
  const float4* kp = (const float4*)(k + (size_t)s * HCC + lane * 8);
  const float4* ep = (const float4*)(ew + lane * 8);
  float4 q0 = qp[0], q1 = qp[1];
  float4 k0 = kp[0], k1 = kp[1];
  float4 e0 = ep[0], e1 = ep[1];

  float dot = q0.x * (k0.x + a * e0.x) + q0.y * (k0.y + a * e0.y)
            + q0.z * (k0.z + a * e0.z) + q0.w * (k0.w + a * e0.w)
            + q1.x * (k1.x + a * e1.x) + q1.y * (k1.y + a * e1.y)
            + q1.z * (k1.z + a * e1.z) + q1.w * (k1.w + a * e1.w);
  // reduce across the 8 lanes that form one head
  dot += __shfl_xor(dot, 1, 32);
  dot += __shfl_xor(dot, 2, 32);
  dot += __shfl_xor(dot, 4, 32);
  if ((lane & 7) == 0) {
    int h = lane >> 3;
    float alpha = dot * 0.125f;    // 1/sqrt(64)
    ealpha[e * HH + h] = alpha;
    atomicMaxF32(&amax[d * HH + h], alpha);
  }
}

// ---------------- attention: exp + segment sum (thread per edge*head) ----------------
__global__ void attn_expsum(float* __restrict__ ealpha, const int* __restrict__ edst,
                            const float* __restrict__ amax, float* __restrict__ asum)
{
  int i = blockIdx.x * 256 + threadIdx.x;   // < NE*HH
  int e = i >> 2, h = i & 3;
  int d = edst[e];
  float ex = expf(ealpha[i] - amax[d * HH + h]);
  ealpha[i] = ex;
  atomicAdd(&asum[d * HH + h], ex);
}

// ---------------- attention: weighted message scatter (wave per edge) ----------------
__global__ __launch_bounds__(128) void attn_scatter(
    const float* __restrict__ v, const int* __restrict__ esrc, const int* __restrict__ edst,
    const float* __restrict__ attr, const float* __restrict__ ew,
    const float* __restrict__ ealpha, const float* __restrict__ asum,
    float* __restrict__ hout)
{
  const int e = blockIdx.x * 4 + (threadIdx.x >> 5);
  const int lane = threadIdx.x & 31;
  const int s = esrc[e], d = edst[e];
  const float a = attr[e];
  const int h = lane >> 3;
  const float w = ealpha[e * HH + h] / (asum[d * HH + h] + 1e-16f);

  const float4* vp = (const float4*)(v + (size_t)s * HCC + lane * 8);
  const float4* ep = (const float4*)(ew + lane * 8);
  float* op = hout + (size_t)d * HCC + lane * 8;
  float4 v0 = vp[0], v1 = vp[1], e0 = ep[0], e1 = ep[1];
  atomicAdd(op + 0, (v0.x + a * e0.x) * w);
  atomicAdd(op + 1, (v0.y + a * e0.y) * w);
  atomicAdd(op + 2, (v0.z + a * e0.z) * w);
  atomicAdd(op + 3, (v0.w + a * e0.w) * w);
  atomicAdd(op + 4, (v1.x + a * e1.x) * w);
  atomicAdd(op + 5, (v1.y + a * e1.y) * w);
  atomicAdd(op + 6, (v1.z + a * e1.z) * w);
  atomicAdd(op + 7, (v1.w + a * e1.w) * w);
}

// ---------------- host-side orchestration ----------------
extern "C" void kernel_launch(void* const* d_in, const int* in_sizes, int n_in,
                              void* d_out, int out_size, void* d_ws, size_t ws_size,
                              hipStream_t stream) {
  const float* x        = (const float*)d_in[0];
  const int*   esrc     = (const int*)  d_in[1];
  const int*   edst     = (const int*)  d_in[2];
  const float* eattr    = (const float*)d_in[3];
  const int*   agent    = (const int*)  d_in[4];
  const float* emb      = (const float*)d_in[5];
  const float* lin1_w   = (const float*)d_in[6];
  const float* lin1_b   = (const float*)d_in[7];
  const float* linh_w   = (const float*)d_in[8];
  const float* linh_b   = (const float*)d_in[9];
  const float* g1_qw = (const float*)d_in[10], *g1_qb = (const float*)d_in[11];
  const float* g1_kw = (const float*)d_in[12], *g1_kb = (const float*)d_in[13];
  const float* g1_vw = (const float*)d_in[14], *g1_vb = (const float*)d_in[15];
  const float* g1_ew = (const float*)d_in[16];
  const float* g1_sw = (const float*)d_in[17], *g1_sb = (const float*)d_in[18];
  const float* g2_qw = (const float*)d_in[19], *g2_qb = (const float*)d_in[20];
  const float* g2_kw = (const float*)d_in[21], *g2_kb = (const float*)d_in[22];
  const float* g2_vw = (const float*)d_in[23], *g2_vb = (const float*)d_in[24];
  const float* g2_ew = (const float*)d_in[25];
  const float* g2_sw = (const float*)d_in[26], *g2_sb = (const float*)d_in[27];
  float* out = (float*)d_out;

  float* ws   = (float*)d_ws;
  float* h0   = ws;                          // NN*128
  float* q    = h0 + (size_t)NN * EH;        // NN*256
  float* k    = q  + (size_t)NN * HCC;
  float* v    = k  + (size_t)NN * HCC;
  float* ha   = v  + (size_t)NN * HCC;       // layer1 out accum
  float* hb   = ha + (size_t)NN * HCC;       // layer2 out accum
  float* amax = hb + (size_t)NN * HCC;       // NN*4
  float* asum = amax + (size_t)NN * HH;      // NN*4
  float* eal  = asum + (size_t)NN * HH;      // NE*4

  // edge MLP -> h0
  fill_kernel<<<(NN * EH) / 256, 256, 0, stream>>>(h0, 0.0f, NN * EH);
  edge_mlp<<<NE / EPB, 128, 0, stream>>>(x, esrc, edst, eattr, emb,
                                         lin1_w, lin1_b, linh_w, linh_b, h0);

  // ---- layer 1: TransformerConv(128 -> 256) ----
  node_qkvs<<<NN / 16, 128, 0, stream>>>(h0, EH,
      g1_qw, g1_qb, g1_kw, g1_kb, g1_vw, g1_vb, g1_sw, g1_sb, q, k, v, ha);
  fill_kernel<<<(NN * HH) / 256, 256, 0, stream>>>(amax, -INFINITY, NN * HH);
  fill_kernel<<<(NN * HH) / 256, 256, 0, stream>>>(asum, 0.0f, NN * HH);
  attn_alpha<<<NE / 4, 128, 0, stream>>>(q, k, esrc, edst, eattr, g1_ew, eal, amax);
  attn_expsum<<<(NE * HH) / 256, 256, 0, stream>>>(eal, edst, amax, asum);
  attn_scatter<<<NE / 4, 128, 0, stream>>>(v, esrc, edst, eattr, g1_ew, eal, asum, ha);
  relu_inplace<<<(NN * HCC) / 256, 256, 0, stream>>>(ha, NN * HCC);

  // ---- layer 2: TransformerConv(256 -> 256) ----
  node_qkvs<<<NN / 16, 128, 0, stream>>>(ha, HCC,
      g2_qw, g2_qb, g2_kw, g2_kb, g2_vw, g2_vb, g2_sw, g2_sb, q, k, v, hb);
  fill_kernel<<<(NN * HH) / 256, 256, 0, stream>>>(amax, -INFINITY, NN * HH);
  fill_kernel<<<(NN * HH) / 256, 256, 0, stream>>>(asum, 0.0f, NN * HH);
  attn_alpha<<<NE / 4, 128, 0, stream>>>(q, k, esrc, edst, eattr, g2_ew, eal, amax);
  attn_expsum<<<(NE * HH) / 256, 256, 0, stream>>>(eal, edst, amax, asum);
  attn_scatter<<<NE / 4, 128, 0, stream>>>(v, esrc, edst, eattr, g2_ew, eal, asum, hb);

  // final relu + 8-row gather
  gather_out<<<NB, HCC, 0, stream>>>(hb, agent, out);
}